// RSLOGIC2Model_6734508720795
// MI455X (gfx1250) — compile-verified
//
#include <hip/hip_runtime.h>

typedef float v2f __attribute__((ext_vector_type(2)));
typedef float v8f __attribute__((ext_vector_type(8)));

#define EPS 1e-40f
#define EMBED_K 64

// ---------------------------------------------------------------------------
// Phase 0: zero per-user log-sum accumulator
// ---------------------------------------------------------------------------
__global__ void zero_f32_kernel(float* __restrict__ p, int n) {
    int i = blockIdx.x * blockDim.x + threadIdx.x;
    if (i < n) p[i] = 0.0f;
}

// ---------------------------------------------------------------------------
// Phase 1: per-edge dot(Gu[u], Gi[i]) via chained V_WMMA_F32_16X16X4_F32.
// One wave handles 16 edges per tile. A = 16 Gu rows (M=edge), B = 16 Gi rows
// (N=edge); diag(D) = the 16 dots. A/B per-lane layout: lane l supplies row
// (l & 15), K components {kb, kb+1} of each 4-wide chunk, kb = 2*(l>>4).
// Diagonal D[m][m]: m<8 -> acc[m] @ lane m ; m>=8 -> acc[m-8] @ lane 16+m.
// ---------------------------------------------------------------------------
__global__ __launch_bounds__(256) void edge_premise_wmma_kernel(
    const float* __restrict__ Gu, const float* __restrict__ Gi,
    const int*  __restrict__ ui, float* __restrict__ logsum,
    int E, int ntiles) {
    const int lane   = threadIdx.x & 31;
    const int wave   = blockIdx.x * (blockDim.x >> 5) + (threadIdx.x >> 5);
    const int nwaves = gridDim.x * (blockDim.x >> 5);
    const int m  = lane & 15;          // edge-in-tile this lane feeds
    const int kb = (lane >> 4) << 1;   // K sub-offset: 0 (lanes 0-15) or 2

    for (int tile = wave; tile < ntiles; tile += nwaves) {
        const int e  = tile * 16 + m;
        const int u  = ui[e];          // row 0 of [2,E]: user ids
        const int it = ui[E + e];      // row 1 of [2,E]: item ids
        const float* pu = Gu + (size_t)u  * EMBED_K + kb;
        const float* pi = Gi + (size_t)it * EMBED_K + kb;

        v8f acc = {};
#pragma unroll
        for (int c = 0; c < 16; ++c) {            // K = 64 in 16 chunks of 4
            v2f a = *(const v2f*)(pu + 4 * c);    // Gu[u][4c+kb .. 4c+kb+1]
            v2f b = *(const v2f*)(pi + 4 * c);    // Gi[i][4c+kb .. 4c+kb+1]
            acc = __builtin_amdgcn_wmma_f32_16x16x4_f32(
                /*neg_a=*/false, a, /*neg_b=*/false, b,
                /*c_mod=*/(short)0, acc, /*reuse_a=*/false, /*reuse_b=*/false);
        }

        // Extract diagonal: lanes 0..7 own edges 0..7, lanes 24..31 own 8..15.
        if (lane < 8 || lane >= 24) {
            const int j = lane & 7;
            float d = acc[0];
            d = (j == 1) ? acc[1] : d;
            d = (j == 2) ? acc[2] : d;
            d = (j == 3) ? acc[3] : d;
            d = (j == 4) ? acc[4] : d;
            d = (j == 5) ? acc[5] : d;
            d = (j == 6) ? acc[6] : d;
            d = (j == 7) ? acc[7] : d;
            // expr = log(sigmoid(d) + eps); segment-sum by user id
            const float s = 1.0f / (1.0f + expf(-d));
            atomicAdd(&logsum[u], logf(s + EPS));
        }
    }
}

// ---------------------------------------------------------------------------
// Phase 1b: scalar tail for E % 16 != 0 (not taken for E = 2,000,000)
// ---------------------------------------------------------------------------
__global__ void edge_premise_tail_kernel(
    const float* __restrict__ Gu, const float* __restrict__ Gi,
    const int* __restrict__ ui, float* __restrict__ logsum,
    int E, int start) {
    int e = start + blockIdx.x * blockDim.x + threadIdx.x;
    if (e >= E) return;
    const int u  = ui[e];
    const int it = ui[E + e];
    const float* pu = Gu + (size_t)u  * EMBED_K;
    const float* pi = Gi + (size_t)it * EMBED_K;
    float d = 0.0f;
#pragma unroll
    for (int k = 0; k < EMBED_K; ++k) d += pu[k] * pi[k];
    const float s = 1.0f / (1.0f + expf(-d));
    atomicAdd(&logsum[u], logf(s + EPS));
}

// ---------------------------------------------------------------------------
// Phase 2: per-batch-element forward logic
// ---------------------------------------------------------------------------
__global__ void finalize_kernel(const int* __restrict__ users,
                                const float* __restrict__ scores,
                                const float* __restrict__ logsum,
                                float* __restrict__ out, int B) {
    int b = blockIdx.x * blockDim.x + threadIdx.x;
    if (b >= B) return;
    const float ls      = logsum[users[b]];
    const float premise = 1.0f + 1.0f / (-1.0f + ls);            // disjunction
    const float s_ui    = 1.0f / (1.0f + expf(-scores[b]));      // sigmoid
    const float ls2 = logf(1.0f - s_ui + EPS) + logf(1.0f - premise + EPS);
    out[b] = 1.0f + 1.0f / (-1.0f + ls2);
}

// ---------------------------------------------------------------------------
extern "C" void kernel_launch(void* const* d_in, const int* in_sizes, int n_in,
                              void* d_out, int out_size, void* d_ws, size_t ws_size,
                              hipStream_t stream) {
    const int*   users  = (const int*)  d_in[0];
    // d_in[1] (items) is unused by the reference
    const float* scores = (const float*)d_in[2];
    const float* Gu     = (const float*)d_in[3];
    const float* Gi     = (const float*)d_in[4];
    const int*   ui     = (const int*)  d_in[5];
    float* out = (float*)d_out;

    const int B = in_sizes[0];
    const int U = in_sizes[3] / EMBED_K;   // NUM_USERS
    const int E = in_sizes[5] / 2;         // edge count

    float* logsum = (float*)d_ws;          // U floats of scratch

    // Phase 0: zero accumulator
    zero_f32_kernel<<<(U + 255) / 256, 256, 0, stream>>>(logsum, U);

    // Phase 1: WMMA edge dots + atomic segment-sum
    const int ntiles = E / 16;
    if (ntiles > 0) {
        const int wavesPerBlock = 8;       // 256 threads = 8 wave32
        int blocks = (ntiles + wavesPerBlock - 1) / wavesPerBlock;
        if (blocks > 8192) blocks = 8192;  // grid-stride over tiles
        edge_premise_wmma_kernel<<<blocks, 256, 0, stream>>>(
            Gu, Gi, ui, logsum, E, ntiles);
    }
    const int tail = E - ntiles * 16;
    if (tail > 0) {
        edge_premise_tail_kernel<<<(tail + 255) / 256, 256, 0, stream>>>(
            Gu, Gi, ui, logsum, E, ntiles * 16);
    }

    // Phase 2: finalize batch outputs
    finalize_kernel<<<(B + 255) / 256, 256, 0, stream>>>(
        users, scores, logsum, out, B);
}